// Proposal_14044543058207
// MI455X (gfx1250) — compile-verified
//
#include <hip/hip_runtime.h>
#include <stdint.h>

// ---------------------------------------------------------------------------
// Faster R-CNN Proposal layer for MI455X (gfx1250, wave32).
//   B=8, A=9, H=W=128 -> 147456 anchors/img; pre-NMS top 3000, post-NMS 300.
// Pipeline: fused score kernel -> 64-bit radix-select + LDS bitonic sort
//           (async-to-LDS staging) -> box gather -> TDM-staged greedy NMS.
// ---------------------------------------------------------------------------

#define AN      9
#define HH      128
#define WW      128
#define NANCH   (AN * HH * WW)          // 147456 anchors per image
#define PRE     3000
#define POST    300
#define NMS_T   0.7f
#define NEGINF  (-1.0e10f)
#define TILE    4096                    // floats per async-staged tile (1024 thr * 4)

typedef unsigned int u32x4 __attribute__((ext_vector_type(4)));
typedef unsigned int u32x8 __attribute__((ext_vector_type(8)));

// Classic 9 anchors: base 16, ratios (0.5,1,2) x scales (8,16,32).
__device__ __constant__ float c_anchors[9][4] = {
    { -84.f,  -40.f,   99.f,   55.f},
    {-176.f,  -88.f,  191.f,  103.f},
    {-360.f, -184.f,  375.f,  199.f},
    { -56.f,  -56.f,   71.f,   71.f},
    {-120.f, -120.f,  135.f,  135.f},
    {-248.f, -248.f,  263.f,  263.f},
    { -36.f,  -80.f,   51.f,   95.f},
    { -80.f, -168.f,   95.f,  183.f},
    {-168.f, -344.f,  183.f,  359.f},
};

// Monotone 64-bit key: (ordered-float << 32) | ~index.
// Bigger key == higher score, ties -> smaller index. Keys are unique.
__device__ __forceinline__ unsigned long long make_key(float s, unsigned int i) {
    unsigned int fb  = __float_as_uint(s);
    unsigned int ord = (fb & 0x80000000u) ? ~fb : (fb | 0x80000000u);
    return ((unsigned long long)ord << 32) | (unsigned long long)(~i);
}

// Recompute clipped proposal box for anchor n of batch b.
__device__ __forceinline__ void compute_box(const float* __restrict__ bbox,
                                            const float* __restrict__ im_info,
                                            int b, unsigned int n, float r[4]) {
    unsigned int a  = n % 9u;
    unsigned int hw = n / 9u;
    unsigned int w  = hw % (unsigned)WW;
    unsigned int h  = hw / (unsigned)WW;
    float sx = 16.f * (float)w, sy = 16.f * (float)h;
    float ax1 = c_anchors[a][0] + sx, ay1 = c_anchors[a][1] + sy;
    float ax2 = c_anchors[a][2] + sx, ay2 = c_anchors[a][3] + sy;

    size_t base = ((size_t)b * (4 * AN) + (size_t)a * 4) * (HH * WW)
                + (size_t)h * WW + w;
    float dx = bbox[base];
    float dy = bbox[base + (size_t)(HH * WW)];
    float dw = bbox[base + (size_t)(2 * HH * WW)];
    float dh = bbox[base + (size_t)(3 * HH * WW)];

    float aw = ax2 - ax1 + 1.f, ah = ay2 - ay1 + 1.f;
    float cx = ax1 + 0.5f * aw, cy = ay1 + 0.5f * ah;
    float pcx = dx * aw + cx,   pcy = dy * ah + cy;
    float pw  = expf(dw) * aw,  ph  = expf(dh) * ah;
    float x1 = pcx - 0.5f * pw, y1 = pcy - 0.5f * ph;
    float x2 = pcx + 0.5f * pw, y2 = pcy + 0.5f * ph;

    float im_h = im_info[b * 3 + 0], im_w = im_info[b * 3 + 1];
    r[0] = fminf(fmaxf(x1, 0.f), im_w - 1.f);
    r[1] = fminf(fmaxf(y1, 0.f), im_h - 1.f);
    r[2] = fminf(fmaxf(x2, 0.f), im_w - 1.f);
    r[3] = fminf(fmaxf(y2, 0.f), im_h - 1.f);
}

// ---------------------------------------------------------------------------
// K1: fused objectness score + min-size filter (boxes recomputed later).
// ---------------------------------------------------------------------------
__global__ void score_kernel(const float* __restrict__ cls,
                             const float* __restrict__ bbox,
                             const float* __restrict__ im_info,
                             float* __restrict__ scores, int batches) {
    int g = blockIdx.x * blockDim.x + threadIdx.x;
    if (g >= batches * NANCH) return;
    int b = g / NANCH;
    unsigned int n = (unsigned int)(g - b * NANCH);
    unsigned int a  = n % 9u;
    unsigned int hw = n / 9u;
    unsigned int w  = hw % (unsigned)WW;
    unsigned int h  = hw / (unsigned)WW;
    size_t soff = ((size_t)b * (2 * AN) + AN + a) * (HH * WW) + (size_t)h * WW + w;
    __builtin_prefetch(cls + soff, 0, 0);      // global_prefetch_b8

    float r[4];
    compute_box(bbox, im_info, b, n, r);
    float ms = 16.f * im_info[b * 3 + 2];
    bool ok = (r[2] - r[0] + 1.f >= ms) && (r[3] - r[1] + 1.f >= ms);
    scores[g] = ok ? cls[soff] : NEGINF;
}

// ---------------------------------------------------------------------------
// K2: per-batch exact top-3000 (64-bit MSB radix select) + LDS bitonic sort.
// Score tiles are staged through LDS with CDNA5 async-to-LDS loads; each lane
// consumes exactly the 16B it issued, so no cross-wave barrier is needed in
// the streaming loop, only s_wait_asynccnt.
// ---------------------------------------------------------------------------
__global__ __launch_bounds__(1024) void selectsort_kernel(
        const float* __restrict__ scores, unsigned int* __restrict__ top_idx) {
    const int b   = blockIdx.x;
    const int tid = threadIdx.x;
    const float* sb = scores + (size_t)b * NANCH;

    __shared__ float              tile[TILE];      // 16 KB staging buffer
    __shared__ unsigned long long lkeys[4096];     // 32 KB sort buffer
    __shared__ unsigned int       hist[256];
    __shared__ unsigned long long sh_prefix;
    __shared__ unsigned int       sh_k, sh_cnt;

    if (tid == 0) { sh_prefix = 0ull; sh_k = PRE; sh_cnt = 0u; }
    __syncthreads();

    const unsigned int lds_off =
        (unsigned int)(uintptr_t)(const void*)&tile[tid * 4];

    // ---- 8-pass MSB-first radix select over 64-bit keys ----
    for (int shift = 56; shift >= 0; shift -= 8) {
        if (tid < 256) hist[tid] = 0u;
        __syncthreads();
        unsigned long long prefix = sh_prefix;

        for (int tb = 0; tb < NANCH; tb += TILE) {
            unsigned long long ga =
                (unsigned long long)(uintptr_t)(sb + tb + tid * 4);
            asm volatile("global_load_async_to_lds_b128 %0, %1, off"
                         :: "v"(lds_off), "v"(ga) : "memory");
            asm volatile("s_wait_asynccnt 0x0" ::: "memory");
            #pragma unroll
            for (int e = 0; e < 4; ++e) {
                unsigned int i = (unsigned int)(tb + tid * 4 + e);
                unsigned long long key = make_key(tile[tid * 4 + e], i);
                bool match = (shift == 56) ||
                             ((key >> (shift + 8)) == (prefix >> (shift + 8)));
                if (match)
                    atomicAdd(&hist[(unsigned int)(key >> shift) & 255u], 1u);
            }
        }
        __syncthreads();
        if (tid == 0) {
            unsigned int k = sh_k;
            int d = 255;
            for (; d > 0; --d) {
                unsigned int c = hist[d];
                if (k <= c) break;
                k -= c;
            }
            sh_prefix = prefix | ((unsigned long long)(unsigned int)d << shift);
            sh_k = k;
        }
        __syncthreads();
    }
    const unsigned long long T = sh_prefix;   // exact 3000th-largest key

    // ---- compaction: key >= T selects exactly PRE elements (keys unique) ----
    for (int i = tid; i < 4096; i += 1024) lkeys[i] = 0ull;  // pad sorts last
    __syncthreads();
    for (int tb = 0; tb < NANCH; tb += TILE) {
        unsigned long long ga =
            (unsigned long long)(uintptr_t)(sb + tb + tid * 4);
        asm volatile("global_load_async_to_lds_b128 %0, %1, off"
                     :: "v"(lds_off), "v"(ga) : "memory");
        asm volatile("s_wait_asynccnt 0x0" ::: "memory");
        #pragma unroll
        for (int e = 0; e < 4; ++e) {
            unsigned int i = (unsigned int)(tb + tid * 4 + e);
            unsigned long long key = make_key(tile[tid * 4 + e], i);
            if (key >= T) {
                unsigned int p = atomicAdd(&sh_cnt, 1u);
                if (p < 4096u) lkeys[p] = key;
            }
        }
    }
    __syncthreads();

    // ---- bitonic sort, descending, 4096 entries in LDS ----
    for (int k = 2; k <= 4096; k <<= 1) {
        for (int j = k >> 1; j > 0; j >>= 1) {
            for (int i = tid; i < 4096; i += 1024) {
                int l = i ^ j;
                if (l > i) {
                    unsigned long long a0 = lkeys[i], a1 = lkeys[l];
                    bool sw = ((i & k) == 0) ? (a0 < a1) : (a0 > a1);
                    if (sw) { lkeys[i] = a1; lkeys[l] = a0; }
                }
            }
            __syncthreads();
        }
    }

    for (int i = tid; i < PRE; i += 1024) {
        unsigned int low = (unsigned int)(lkeys[i] & 0xFFFFFFFFull);
        top_idx[(size_t)b * PRE + i] = ~low;   // recover anchor index
    }
}

// ---------------------------------------------------------------------------
// K3: gather (recompute) the clipped boxes of the 3000 winners per batch.
// ---------------------------------------------------------------------------
__global__ void gather_kernel(const float* __restrict__ bbox,
                              const float* __restrict__ im_info,
                              const unsigned int* __restrict__ top_idx,
                              float* __restrict__ topboxes, int batches) {
    int g = blockIdx.x * blockDim.x + threadIdx.x;
    if (g >= batches * PRE) return;
    int b = g / PRE;
    unsigned int n = top_idx[g];
    float r[4];
    compute_box(bbox, im_info, b, n, r);
    topboxes[(size_t)g * 4 + 0] = r[0];
    topboxes[(size_t)g * 4 + 1] = r[1];
    topboxes[(size_t)g * 4 + 2] = r[2];
    topboxes[(size_t)g * 4 + 3] = r[3];
}

// ---------------------------------------------------------------------------
// K4: greedy NMS + post-NMS selection. The 48 KB sorted-box tile is pulled
// into LDS by the Tensor Data Mover (one 2D 4x3000 descriptor per workgroup),
// synchronized with s_wait_tensorcnt.
// ---------------------------------------------------------------------------
__global__ __launch_bounds__(1024) void nms_kernel(
        const float* __restrict__ topboxes, float* __restrict__ rois) {
    const int b   = blockIdx.x;
    const int tid = threadIdx.x;
    __shared__ float        box[PRE][4];    // 48 KB
    __shared__ unsigned int keep[PRE];      // 12 KB

    const float* src = topboxes + (size_t)b * PRE * 4;

    if (tid < 32) {   // wave 0 issues the TDM descriptor (EXEC ignored by TDM)
        unsigned long long ga = (unsigned long long)(uintptr_t)(const void*)src;
        unsigned int lds = (unsigned int)(uintptr_t)(void*)&box[0][0];
        u32x4 g0; u32x8 g1; u32x4 gz = (u32x4)(0u);
        g0.x = 1u;                                           // count=1, load-descriptor
        g0.y = lds;                                          // LDS byte address
        g0.z = (unsigned int)(ga & 0xFFFFFFFFull);           // global addr [95:64]
        g0.w = (unsigned int)((ga >> 32) & 0x01FFFFFFull)    // global addr [120:96]
             | (2u << 30);                                   // type = 2 (image)
        g1.s0 = (2u << 16);          // wg_mask=0, data_size=2 (4 bytes)
        g1.s1 = (4u    << 16);       // tensor_dim0 = 4   (bits 79:48, low half)
        g1.s2 = (3000u << 16);       // tensor_dim0 hi | tensor_dim1 = 3000 (low)
        g1.s3 = (4u    << 16);       // tensor_dim1 hi | tile_dim0 = 4
        g1.s4 = 3000u;               // tile_dim1 = 3000, tile_dim2 = 0
        g1.s5 = 4u;                  // tensor_dim0_stride = 4 elements
        g1.s6 = 0u;                  // stride0 hi | stride1 lo
        g1.s7 = 0u;                  // stride1 hi
        asm volatile("tensor_load_to_lds %0, %1, %2, %3"
                     :: "s"(g0), "s"(g1), "s"(gz), "s"(gz) : "memory");
    }
    __builtin_amdgcn_s_wait_tensorcnt(0);   // no-op for non-issuing waves
    for (int j = tid; j < PRE; j += 1024) keep[j] = 1u;
    __syncthreads();

    // Greedy NMS: sequential over rows, parallel over columns.
    for (int i = 0; i < PRE; ++i) {
        __syncthreads();
        if (keep[i] == 0u) continue;        // uniform branch
        float x1i = box[i][0], y1i = box[i][1], x2i = box[i][2], y2i = box[i][3];
        float ai  = (x2i - x1i + 1.f) * (y2i - y1i + 1.f);
        for (int j = i + 1 + tid; j < PRE; j += 1024) {
            if (!keep[j]) continue;
            float xx1 = fmaxf(x1i, box[j][0]);
            float yy1 = fmaxf(y1i, box[j][1]);
            float xx2 = fminf(x2i, box[j][2]);
            float yy2 = fminf(y2i, box[j][3]);
            float iw  = fmaxf(xx2 - xx1 + 1.f, 0.f);
            float ih  = fmaxf(yy2 - yy1 + 1.f, 0.f);
            float inter = iw * ih;
            float aj = (box[j][2] - box[j][0] + 1.f) * (box[j][3] - box[j][1] + 1.f);
            if (inter > NMS_T * (ai + aj - inter)) keep[j] = 0u;
        }
    }
    __syncthreads();

    // Post-NMS top-300: kept boxes in score order, then suppressed (NEG_INF
    // ties resolve by index in jax top_k) as pad.
    if (tid == 0) {
        float* out = rois + (size_t)b * POST * 4;
        int cnt = 0;
        for (int j = 0; j < PRE && cnt < POST; ++j)
            if (keep[j]) {
                out[cnt * 4 + 0] = box[j][0]; out[cnt * 4 + 1] = box[j][1];
                out[cnt * 4 + 2] = box[j][2]; out[cnt * 4 + 3] = box[j][3];
                ++cnt;
            }
        for (int j = 0; j < PRE && cnt < POST; ++j)
            if (!keep[j]) {
                out[cnt * 4 + 0] = box[j][0]; out[cnt * 4 + 1] = box[j][1];
                out[cnt * 4 + 2] = box[j][2]; out[cnt * 4 + 3] = box[j][3];
                ++cnt;
            }
    }
}

// ---------------------------------------------------------------------------
extern "C" void kernel_launch(void* const* d_in, const int* in_sizes, int n_in,
                              void* d_out, int out_size, void* d_ws, size_t ws_size,
                              hipStream_t stream) {
    (void)n_in; (void)out_size;
    const float* cls     = (const float*)d_in[0];   // (B, 18, 128, 128)
    const float* bbox    = (const float*)d_in[1];   // (B, 36, 128, 128)
    const float* im_info = (const float*)d_in[2];   // (B, 3)
    const int batches = in_sizes[2] / 3;            // B = 8

    // Workspace layout (256B aligned): scores | top_idx | topboxes
    size_t sz_scores = (size_t)batches * NANCH * sizeof(float);
    size_t off_idx   = (sz_scores + 255) & ~(size_t)255;
    size_t sz_idx    = (size_t)batches * PRE * sizeof(unsigned int);
    size_t off_boxes = ((off_idx + sz_idx) + 255) & ~(size_t)255;
    size_t sz_boxes  = (size_t)batches * PRE * 4 * sizeof(float);
    if (ws_size < off_boxes + sz_boxes) return;

    float*        scores   = (float*)d_ws;
    unsigned int* top_idx  = (unsigned int*)((char*)d_ws + off_idx);
    float*        topboxes = (float*)((char*)d_ws + off_boxes);
    float*        rois     = (float*)d_out;

    int total = batches * NANCH;
    score_kernel<<<(total + 255) / 256, 256, 0, stream>>>(
        cls, bbox, im_info, scores, batches);
    selectsort_kernel<<<batches, 1024, 0, stream>>>(scores, top_idx);
    gather_kernel<<<(batches * PRE + 255) / 256, 256, 0, stream>>>(
        bbox, im_info, top_idx, topboxes, batches);
    nms_kernel<<<batches, 1024, 0, stream>>>(topboxes, rois);
}